// SpatialPyramidParameters_4380866642085
// MI455X (gfx1250) — compile-verified
//
#include <hip/hip_runtime.h>

// out[b, :] = sum_h params_h[grid[h, loc_b], t_b, :]
//   H=5 levels, T=32, K=128, L=100000, B=500000.
// Bandwidth-bound: ~256 MB streamed output + ~11 MB table/index traffic.
// One wave32 per batch element: lane l holds 4 floats = out[b, 4l:4l+4].
// Index chain is wave-uniform -> force SGPR via readfirstlane.
// Output store is non-temporal (TH=NT) so the 256 MB stream doesn't evict
// the 5.3 MB of hot parameter tables (or the 2 MB grid table) from L2.

#define T_DIM 32
#define K_DIM 128
#define WAVES_PER_BLOCK 8

typedef float v4f __attribute__((ext_vector_type(4)));

__global__ __launch_bounds__(256, 1) void pyr_gather_sum_kernel(
    const float* __restrict__ p0,
    const float* __restrict__ p1,
    const float* __restrict__ p2,
    const float* __restrict__ p3,
    const float* __restrict__ p4,
    const int*   __restrict__ grid_assign,   // [5, L]
    const int*   __restrict__ loc_idx,       // [B]
    const int*   __restrict__ t_idx,         // [B]
    float*       __restrict__ out,           // [B, 128]
    int L, int B)
{
    const int lane = threadIdx.x & 31;
    int b = blockIdx.x * WAVES_PER_BLOCK + (threadIdx.x >> 5);
    if (b >= B) return;                 // whole wave exits together (b is per-wave)

    // Wave-uniform index chain -> keep it scalar (SGPR / s_load path).
    const int bu  = __builtin_amdgcn_readfirstlane(b);
    const int loc = __builtin_amdgcn_readfirstlane(loc_idx[bu]);
    const int t   = __builtin_amdgcn_readfirstlane(t_idx[bu]);

    const int c0 = __builtin_amdgcn_readfirstlane(grid_assign[0 * L + loc]);
    const int c1 = __builtin_amdgcn_readfirstlane(grid_assign[1 * L + loc]);
    const int c2 = __builtin_amdgcn_readfirstlane(grid_assign[2 * L + loc]);
    const int c3 = __builtin_amdgcn_readfirstlane(grid_assign[3 * L + loc]);
    const int c4 = __builtin_amdgcn_readfirstlane(grid_assign[4 * L + loc]);

    // Five coalesced 512 B row reads (global_load_b128 per lane).
    const v4f* r0 = (const v4f*)(p0 + (size_t)(c0 * T_DIM + t) * K_DIM);
    const v4f* r1 = (const v4f*)(p1 + (size_t)(c1 * T_DIM + t) * K_DIM);
    const v4f* r2 = (const v4f*)(p2 + (size_t)(c2 * T_DIM + t) * K_DIM);
    const v4f* r3 = (const v4f*)(p3 + (size_t)(c3 * T_DIM + t) * K_DIM);
    const v4f* r4 = (const v4f*)(p4 + (size_t)(c4 * T_DIM + t) * K_DIM);

    v4f v0 = r0[lane];
    v4f v1 = r1[lane];
    v4f v2 = r2[lane];
    v4f v3 = r3[lane];
    v4f v4 = r4[lane];

    v4f acc = v0 + v1 + v2 + v3 + v4;

    // Streamed output: non-temporal b128 store to preserve L2 for tables.
    v4f* o = (v4f*)(out + (size_t)b * K_DIM);
    __builtin_nontemporal_store(acc, o + lane);
}

extern "C" void kernel_launch(void* const* d_in, const int* in_sizes, int n_in,
                              void* d_out, int out_size, void* d_ws, size_t ws_size,
                              hipStream_t stream) {
    (void)n_in; (void)out_size; (void)d_ws; (void)ws_size;

    const float* p0 = (const float*)d_in[0];
    const float* p1 = (const float*)d_in[1];
    const float* p2 = (const float*)d_in[2];
    const float* p3 = (const float*)d_in[3];
    const float* p4 = (const float*)d_in[4];
    const int* grid_assign = (const int*)d_in[5];
    const int* loc_idx     = (const int*)d_in[6];
    const int* t_idx       = (const int*)d_in[7];
    float* out = (float*)d_out;

    const int L = in_sizes[5] / 5;   // grid_assignments is [5, L]
    const int B = in_sizes[6];       // location_indices is [B]

    const int blocks = (B + WAVES_PER_BLOCK - 1) / WAVES_PER_BLOCK;
    pyr_gather_sum_kernel<<<blocks, WAVES_PER_BLOCK * 32, 0, stream>>>(
        p0, p1, p2, p3, p4, grid_assign, loc_idx, t_idx, out, L, B);
}